// GAT_LSTM_22368189678360
// MI455X (gfx1250) — compile-verified
//
#include <hip/hip_runtime.h>
#include <hip/hip_bf16.h>
#include <math.h>

#define GG 80          // S*T graphs
#define NN 2000
#define EE 16000
#define AE (EE + NN)   // edges + self loops
#define LH 256         // LSTM hidden
#define G4 1024        // 4*LH
#define SEQD 16000     // N*TGT
#define NEG 0.2f

typedef float v2f __attribute__((ext_vector_type(2)));
typedef float v8f __attribute__((ext_vector_type(8)));

__device__ __forceinline__ unsigned fkey(float f) {
  unsigned u = __float_as_uint(f);
  return (u & 0x80000000u) ? ~u : (u | 0x80000000u);
}
__device__ __forceinline__ float funkey(unsigned k) {
  unsigned u = (k & 0x80000000u) ? (k ^ 0x80000000u) : ~k;
  return __uint_as_float(u);
}
__device__ __forceinline__ float sigmf(float x) { return 1.0f / (1.0f + __expf(-x)); }

// ---------------------------------------------------------------------------
// WMMA f32 16x16x4 GEMM, C = A[M,K] * B[K,Nd], B row-major [K,Nd].
// One wave per 16x16 output tile; Nd may be < 16 (layer 2: dout=8) -> clamp.
// ---------------------------------------------------------------------------
__global__ void __launch_bounds__(128) gemm16_nn(
    const float* __restrict__ A, const float* __restrict__ B,
    float* __restrict__ C, int K, int Nd, int ntiles, int totTiles)
{
  int tile = blockIdx.x * (blockDim.x >> 5) + (threadIdx.x >> 5);
  if (tile >= totTiles) return;
  int lane = threadIdx.x & 31;
  int half = lane >> 4, l16 = lane & 15;
  int mt = tile / ntiles, nt = tile - mt * ntiles;
  int m0 = mt << 4, n0 = nt << 4;
  int ncol = n0 + l16;
  bool nok = ncol < Nd;
  int ncl = nok ? ncol : (Nd - 1);
  const float* arow = A + (size_t)(m0 + l16) * K;
  v8f acc = {};
  for (int k = 0; k < K; k += 4) {
    int ka = k + 2 * half;
    v2f a = *(const v2f*)(arow + ka);            // A[M=l16, ka..ka+1]
    float b0 = B[(size_t)ka * Nd + ncl];
    float b1 = B[(size_t)(ka + 1) * Nd + ncl];
    v2f b; b.x = nok ? b0 : 0.0f; b.y = nok ? b1 : 0.0f;
    acc = __builtin_amdgcn_wmma_f32_16x16x4_f32(false, a, false, b,
                                                (short)0, acc, false, false);
  }
#pragma unroll
  for (int i = 0; i < 8; ++i)
    if (nok) C[(size_t)(m0 + i + 8 * half) * Nd + ncol] = acc[i];
}

// ---------------------------------------------------------------------------
// WMMA f32 16x16x4 NT GEMM with split-K atomic reduction:
// C[M,Nd] += A[M,K] * W[Nd,K]^T   (dot along contiguous K of both operands)
// ---------------------------------------------------------------------------
__global__ void __launch_bounds__(128) gemm16_nt_splitk(
    const float* __restrict__ A, const float* __restrict__ W,
    float* __restrict__ C, int K, int Nd, int ntiles, int totTiles, int klen)
{
  int tile = blockIdx.x * (blockDim.x >> 5) + (threadIdx.x >> 5);
  if (tile >= totTiles) return;
  int lane = threadIdx.x & 31;
  int half = lane >> 4, l16 = lane & 15;
  int mt = tile / ntiles, nt = tile - mt * ntiles;
  int m0 = mt << 4, n0 = nt << 4;
  const float* arow = A + (size_t)(m0 + l16) * K;
  const float* wrow = W + (size_t)(n0 + l16) * K;
  int k0 = blockIdx.y * klen;
  v8f acc = {};
  for (int k = k0; k < k0 + klen; k += 4) {
    int ka = k + 2 * half;
    v2f a = *(const v2f*)(arow + ka);
    v2f b = *(const v2f*)(wrow + ka);
    acc = __builtin_amdgcn_wmma_f32_16x16x4_f32(false, a, false, b,
                                                (short)0, acc, false, false);
  }
#pragma unroll
  for (int i = 0; i < 8; ++i)
    atomicAdd(&C[(size_t)(m0 + i + 8 * half) * Nd + (n0 + l16)], acc[i]);
}

// --------------------------- GAT support kernels ---------------------------
__global__ void count_k(const int* __restrict__ ei, float* __restrict__ cnt) {
  int e = blockIdx.x * blockDim.x + threadIdx.x;
  if (e < EE) atomicAdd(&cnt[ei[EE + e]], 1.0f);
}

__global__ void loopea_acc_k(const int* __restrict__ ei, const float* __restrict__ ea,
                             float* __restrict__ lea) {
  int idx = blockIdx.x * blockDim.x + threadIdx.x;
  if (idx >= GG * EE) return;
  int g = idx / EE, e = idx - g * EE;
  int d = ei[EE + e];
  const float* p = ea + ((size_t)g * EE + e) * 2;
  atomicAdd(&lea[((size_t)g * NN + d) * 2 + 0], p[0]);
  atomicAdd(&lea[((size_t)g * NN + d) * 2 + 1], p[1]);
}

__global__ void loopea_div_k(const float* __restrict__ cnt, float* __restrict__ lea) {
  int idx = blockIdx.x * blockDim.x + threadIdx.x;
  if (idx >= GG * NN) return;
  float c = fmaxf(cnt[idx % NN], 1.0f);
  lea[(size_t)idx * 2 + 0] /= c;
  lea[(size_t)idx * 2 + 1] /= c;
}

__global__ void edge_logit_k(const float* __restrict__ hl, const float* __restrict__ hr,
                             const int* __restrict__ ei, const float* __restrict__ ea,
                             const float* __restrict__ lea, const float* __restrict__ We,
                             const float* __restrict__ att, float* __restrict__ lg,
                             unsigned* __restrict__ lmx, int dout)
{
  int idx = blockIdx.x * blockDim.x + threadIdx.x;
  if (idx >= GG * AE) return;
  int g = idx / AE, a = idx - g * AE;
  int s, d; float e0, e1;
  if (a < EE) {
    s = ei[a]; d = ei[EE + a];
    const float* p = ea + ((size_t)g * EE + a) * 2;
    e0 = p[0]; e1 = p[1];
  } else {
    s = d = a - EE;
    const float* p = lea + ((size_t)g * NN + s) * 2;
    e0 = p[0]; e1 = p[1];
  }
  const float* pl = hl + ((size_t)g * NN + s) * dout;
  const float* pr = hr + ((size_t)g * NN + d) * dout;
  float acc = 0.0f;
#pragma unroll 8
  for (int k = 0; k < dout; ++k) {
    float m = pl[k] + pr[k] + e0 * We[k] + e1 * We[dout + k];
    m = (m > 0.0f) ? m : NEG * m;   // leaky_relu
    acc += m * att[k];
  }
  lg[idx] = acc;
  atomicMax(&lmx[g * NN + d], fkey(acc));
}

__global__ void edge_exp_k(float* __restrict__ lg, const unsigned* __restrict__ lmx,
                           float* __restrict__ den, const int* __restrict__ ei)
{
  int idx = blockIdx.x * blockDim.x + threadIdx.x;
  if (idx >= GG * AE) return;
  int g = idx / AE, a = idx - g * AE;
  int d = (a < EE) ? ei[EE + a] : (a - EE);
  float ex = __expf(lg[idx] - funkey(lmx[g * NN + d]));
  lg[idx] = ex;                      // reuse buffer for exp values
  atomicAdd(&den[g * NN + d], ex);
}

__global__ void outinit_k(float* __restrict__ out, const float* __restrict__ b, int dout,
                          long total) {
  long idx = (long)blockIdx.x * blockDim.x + threadIdx.x;
  if (idx < total) out[idx] = b[idx % dout];
}

__global__ void aggregate_k(const float* __restrict__ hl, const float* __restrict__ ex,
                            const float* __restrict__ den, const int* __restrict__ ei,
                            float* __restrict__ out, int dout, int per)
{
  long idx = (long)blockIdx.x * blockDim.x + threadIdx.x;
  if (idx >= (long)GG * AE * per) return;
  int kc = idx % per;
  int a  = (idx / per) % AE;
  int g  = idx / ((long)per * AE);
  int s, d;
  if (a < EE) { s = ei[a]; d = ei[EE + a]; } else { s = d = a - EE; }
  float w = ex[(size_t)g * AE + a] / den[g * NN + d];
  const float* src = hl + ((size_t)g * NN + s) * dout;
  float* dst = out + ((size_t)g * NN + d) * dout;
  int k0 = kc * 16, k1 = min(k0 + 16, dout);
  for (int k = k0; k < k1; ++k) atomicAdd(&dst[k], w * src[k]);
}

// ----------------------------- LSTM kernels --------------------------------
__global__ void ginit_k(float* __restrict__ G, const float* __restrict__ b0,
                        const float* __restrict__ b1) {
  int idx = blockIdx.x * blockDim.x + threadIdx.x;
  if (idx < GG * G4) { int j = idx % G4; G[idx] = b0[j] + b1[j]; }
}

__global__ void __launch_bounds__(1024) lstm_step_k(
    const float* __restrict__ G0, const float* __restrict__ Whh0,
    const float* __restrict__ Wih1, const float* __restrict__ Whh1,
    const float* __restrict__ bih1, const float* __restrict__ bhh1,
    float* __restrict__ h0, float* __restrict__ c0,
    float* __restrict__ h1, float* __restrict__ c1, int t)
{
  __shared__ float sh0[4][LH];
  __shared__ float sh1[4][LH];
  __shared__ float gate[4][G4];
  int j = threadIdx.x;                      // 0..1023
  if (j < LH)
    for (int s = 0; s < 4; ++s) { sh0[s][j] = h0[s * LH + j]; sh1[s][j] = h1[s * LH + j]; }
  __syncthreads();

  // layer 0 gates: precomputed input part + recurrent part
  {
    const float* wr = Whh0 + (size_t)j * LH;
    for (int s = 0; s < 4; ++s) {
      float acc = G0[(size_t)(s * 20 + t) * G4 + j];
#pragma unroll 4
      for (int k = 0; k < LH; ++k) acc += wr[k] * sh0[s][k];
      gate[s][j] = acc;
    }
  }
  __syncthreads();
  if (j < LH) {
    for (int s = 0; s < 4; ++s) {
      float ig = gate[s][j], fg = gate[s][LH + j];
      float gg = gate[s][2 * LH + j], og = gate[s][3 * LH + j];
      float c = sigmf(fg) * c0[s * LH + j] + sigmf(ig) * tanhf(gg);
      float h = sigmf(og) * tanhf(c);
      c0[s * LH + j] = c; h0[s * LH + j] = h; sh0[s][j] = h;
    }
  }
  __syncthreads();

  // layer 1 gates: input = fresh layer-0 h
  {
    const float* wi = Wih1 + (size_t)j * LH;
    const float* wh = Whh1 + (size_t)j * LH;
    float bb = bih1[j] + bhh1[j];
    for (int s = 0; s < 4; ++s) {
      float acc = bb;
#pragma unroll 4
      for (int k = 0; k < LH; ++k) acc += wi[k] * sh0[s][k] + wh[k] * sh1[s][k];
      gate[s][j] = acc;
    }
  }
  __syncthreads();
  if (j < LH) {
    for (int s = 0; s < 4; ++s) {
      float ig = gate[s][j], fg = gate[s][LH + j];
      float gg = gate[s][2 * LH + j], og = gate[s][3 * LH + j];
      float c = sigmf(fg) * c1[s * LH + j] + sigmf(ig) * tanhf(gg);
      float h = sigmf(og) * tanhf(c);
      c1[s * LH + j] = c; h1[s * LH + j] = h;
    }
  }
}

__global__ void fc_k(const float* __restrict__ h1, const float* __restrict__ fcW,
                     const float* __restrict__ fcb, float* __restrict__ out) {
  __shared__ float red[LH];
  int s = blockIdx.x, j = threadIdx.x;
  red[j] = h1[s * LH + j] * fcW[j];
  __syncthreads();
  for (int off = LH / 2; off > 0; off >>= 1) {
    if (j < off) red[j] += red[j + off];
    __syncthreads();
  }
  if (j == 0) out[s] = red[0] + fcb[0];
}

// ---------------------------------------------------------------------------
extern "C" void kernel_launch(void* const* d_in, const int* in_sizes, int n_in,
                              void* d_out, int out_size, void* d_ws, size_t ws_size,
                              hipStream_t stream)
{
  (void)in_sizes; (void)n_in; (void)out_size; (void)ws_size;
  const float* x   = (const float*)d_in[0];
  const int*   ei  = (const int*)d_in[1];
  const float* ea  = (const float*)d_in[2];
  const float* Wl[3] = {(const float*)d_in[3], (const float*)d_in[8],  (const float*)d_in[13]};
  const float* Wr[3] = {(const float*)d_in[4], (const float*)d_in[9],  (const float*)d_in[14]};
  const float* We[3] = {(const float*)d_in[5], (const float*)d_in[10], (const float*)d_in[15]};
  const float* at[3] = {(const float*)d_in[6], (const float*)d_in[11], (const float*)d_in[16]};
  const float* bg[3] = {(const float*)d_in[7], (const float*)d_in[12], (const float*)d_in[17]};
  const float* Wih0 = (const float*)d_in[18];
  const float* Whh0 = (const float*)d_in[19];
  const float* bih0 = (const float*)d_in[20];
  const float* bhh0 = (const float*)d_in[21];
  const float* Wih1 = (const float*)d_in[22];
  const float* Whh1 = (const float*)d_in[23];
  const float* bih1 = (const float*)d_in[24];
  const float* bhh1 = (const float*)d_in[25];
  const float* fcW  = (const float*)d_in[26];
  const float* fcb  = (const float*)d_in[27];

  float* base = (float*)d_ws;
  size_t o = 0;
  auto take = [&](size_t n) { float* p = base + o; o += (n + 255) & ~(size_t)255; return p; };
  float*    cnt = take(NN);
  float*    lea = take((size_t)GG * NN * 2);
  float*    hA  = take((size_t)GG * NN * 64);
  float*    hB  = take((size_t)GG * NN * 64);
  float*    hl  = take((size_t)GG * NN * 64);
  float*    hr  = take((size_t)GG * NN * 64);
  float*    lg  = take((size_t)GG * AE);
  unsigned* lmx = (unsigned*)take((size_t)GG * NN);
  float*    den = take((size_t)GG * NN);
  float*    Gb  = take((size_t)GG * G4);
  float*    st  = take(16 * LH);                 // h0,c0,h1,c1 states
  float *h0s = st, *c0s = st + 4 * LH, *h1s = st + 8 * LH, *c1s = st + 12 * LH;

  hipMemsetAsync(cnt, 0, NN * sizeof(float), stream);
  hipMemsetAsync(lea, 0, (size_t)GG * NN * 2 * sizeof(float), stream);
  hipMemsetAsync(st,  0, 16 * LH * sizeof(float), stream);

  // degree counts + self-loop edge-attr mean (per graph)
  count_k<<<(EE + 255) / 256, 256, 0, stream>>>(ei, cnt);
  loopea_acc_k<<<(GG * EE + 255) / 256, 256, 0, stream>>>(ei, ea, lea);
  loopea_div_k<<<(GG * NN + 255) / 256, 256, 0, stream>>>(cnt, lea);

  // ---- 3 GAT layers, all 80 graphs batched ----
  const float* lin[3]  = {x, hA, hB};
  float*       lout[3] = {hA, hB, hA};
  const int din[3]  = {16, 64, 64};
  const int dcol[3] = {64, 64, 8};
  for (int l = 0; l < 3; ++l) {
    int ntiles = (dcol[l] + 15) / 16;
    int tot = (GG * NN / 16) * ntiles;           // M tiles exact: 160000/16
    int blocks = (tot + 3) / 4;                  // 4 waves / block
    gemm16_nn<<<blocks, 128, 0, stream>>>(lin[l], Wl[l], hl, din[l], dcol[l], ntiles, tot);
    gemm16_nn<<<blocks, 128, 0, stream>>>(lin[l], Wr[l], hr, din[l], dcol[l], ntiles, tot);
    hipMemsetAsync(lmx, 0, (size_t)GG * NN * 4, stream);   // 0 == -inf key floor
    hipMemsetAsync(den, 0, (size_t)GG * NN * 4, stream);
    edge_logit_k<<<(GG * AE + 255) / 256, 256, 0, stream>>>(hl, hr, ei, ea, lea,
                                                            We[l], at[l], lg, lmx, dcol[l]);
    edge_exp_k<<<(GG * AE + 255) / 256, 256, 0, stream>>>(lg, lmx, den, ei);
    long totOut = (long)GG * NN * dcol[l];
    outinit_k<<<(unsigned)((totOut + 255) / 256), 256, 0, stream>>>(lout[l], bg[l], dcol[l], totOut);
    int per = ntiles;
    long totAg = (long)GG * AE * per;
    aggregate_k<<<(unsigned)((totAg + 255) / 256), 256, 0, stream>>>(hl, lg, den, ei,
                                                                    lout[l], dcol[l], per);
  }
  // final embeddings in hA: [80][N*TGT] == seq[S,T,16000]

  // ---- LSTM input projection for ALL timesteps at once (stream Wih0 once) ----
  ginit_k<<<(GG * G4 + 255) / 256, 256, 0, stream>>>(Gb, bih0, bhh0);
  int tot2 = (GG / 16) * (G4 / 16);              // 5 * 64 = 320 tiles
  gemm16_nt_splitk<<<dim3((tot2 + 3) / 4, 16), 128, 0, stream>>>(
      hA, Wih0, Gb, SEQD, G4, G4 / 16, tot2, SEQD / 16);

  // ---- sequential recurrence: one fused (2-layer) step kernel per t ----
  for (int t = 0; t < 20; ++t)
    lstm_step_k<<<1, 1024, 0, stream>>>(Gb, Whh0, Wih1, Whh1, bih1, bhh1,
                                        h0s, c0s, h1s, c1s, t);

  fc_k<<<4, LH, 0, stream>>>(h1s, fcW, fcb, (float*)d_out);
}